// GraphSAGEEncoder_25426206392892
// MI455X (gfx1250) — compile-verified
//
#include <hip/hip_runtime.h>

typedef __attribute__((ext_vector_type(2))) float v2f;
typedef __attribute__((ext_vector_type(8))) float v8f;

// ---------------------------------------------------------------------------
// Zero a float buffer (grid-stride).
// ---------------------------------------------------------------------------
__global__ void zero_f32(float* __restrict__ p, long n) {
    long i = (long)blockIdx.x * blockDim.x + threadIdx.x;
    long stride = (long)gridDim.x * blockDim.x;
    for (; i < n; i += stride) p[i] = 0.0f;
}

// ---------------------------------------------------------------------------
// Transpose a K x NO row-major weight matrix into NO x K row-major (i.e.
// column-major of the original), so the GEMM's B operand becomes a contiguous
// float2 load per lane. One-time cost, 8K elements per matrix.
// ---------------------------------------------------------------------------
__global__ void transpose_w(const float* __restrict__ in, float* __restrict__ outp,
                            int K, int NO) {
    int i = blockIdx.x * blockDim.x + threadIdx.x;
    if (i < K * NO) {
        int k = i / NO;
        int n = i % NO;
        outp[n * K + k] = in[i];
    }
}

// ---------------------------------------------------------------------------
// Layer-1 aggregation: one wave per edge, F = 64 (each lane moves 2 floats).
// Also accumulates per-destination edge counts (lane 0).
// ---------------------------------------------------------------------------
__global__ __launch_bounds__(256)
void agg_f64(const float* __restrict__ x, const long long* __restrict__ srcs,
             const long long* __restrict__ dsts, float* __restrict__ agg,
             float* __restrict__ cnt, int E) {
    int wave = threadIdx.x >> 5;
    int lane = threadIdx.x & 31;
    int e = blockIdx.x * 8 + wave;
    if (e >= E) return;                         // wave-uniform branch
    long s = (long)srcs[e];
    long d = (long)dsts[e];
    const v2f* xr = (const v2f*)(x + s * 64);
    v2f v = xr[lane];
    float* ad = agg + d * 64 + lane * 2;
    atomicAdd(ad + 0, v.x);
    atomicAdd(ad + 1, v.y);
    if (lane == 0) atomicAdd(cnt + d, 1.0f);
}

// ---------------------------------------------------------------------------
// Layer-2 aggregation: one wave per edge, F = 128 (each lane moves 4 floats).
// ---------------------------------------------------------------------------
__global__ __launch_bounds__(256)
void agg_f128(const float* __restrict__ h, const long long* __restrict__ srcs,
              const long long* __restrict__ dsts, float* __restrict__ agg, int E) {
    int wave = threadIdx.x >> 5;
    int lane = threadIdx.x & 31;
    int e = blockIdx.x * 8 + wave;
    if (e >= E) return;                         // wave-uniform branch
    long s = (long)srcs[e];
    long d = (long)dsts[e];
    const float4* hr = (const float4*)(h + s * 128);
    float4 v = hr[lane];
    float* ad = agg + d * 128 + lane * 4;
    atomicAdd(ad + 0, v.x);
    atomicAdd(ad + 1, v.y);
    atomicAdd(ad + 2, v.z);
    atomicAdd(ad + 3, v.w);
}

// ---------------------------------------------------------------------------
// Fused SAGE dense stage:  out = act( (Agg/max(cnt,1)) @ Wl  +  X @ Wr  + b )
// One wave per 16x16 output tile, V_WMMA_F32_16X16X4_F32, fully unrolled.
//
// fp32 WMMA operand layout (ISA 7.12.2):
//   A (16x4):  lane L -> row m = L&15,  float2 covers K = 2*(L>>4)+{0,1}
//   B (4x16):  lane L -> col n = L&15,  float2 covers K = 2*(L>>4)+{0,1}
//   C/D:       lane L -> col n = L&15,  vgpr r -> row m = r + 8*(L>>4)
//
// WlT / WrT are NO x K (pre-transposed) so B is a contiguous b64 load.
// Grid is exact (N % 16 == 0): no bounds checks, EXEC stays all-1s.
// ---------------------------------------------------------------------------
template <int K, int NO, bool RELU>
__global__ __launch_bounds__(256)
void sage_gemm(const float* __restrict__ Agg, const float* __restrict__ cnt,
               const float* __restrict__ X, const float* __restrict__ WlT,
               const float* __restrict__ WrT, const float* __restrict__ bias,
               float* __restrict__ out) {
    constexpr int TN = NO / 16;                 // tiles along output columns
    int wave = threadIdx.x >> 5;
    int lane = threadIdx.x & 31;
    int tile = blockIdx.x * 8 + wave;
    int tm = tile / TN;
    int tn = tile % TN;
    int row  = lane & 15;
    int half = lane >> 4;

    int m = tm * 16 + row;                      // A row this lane feeds
    int n = tn * 16 + row;                      // B/C/D column this lane feeds

    float inv = 1.0f / fmaxf(cnt[m], 1.0f);     // mean scale, fixed per lane

    const v2f* arow = (const v2f*)(Agg + (long)m * K);
    const v2f* xrow = (const v2f*)(X   + (long)m * K);
    const v2f* blp  = (const v2f*)(WlT + (long)n * K);
    const v2f* brp  = (const v2f*)(WrT + (long)n * K);

    v8f acc = {};

    // ---- mean @ Wl -------------------------------------------------------
#pragma unroll
    for (int k0 = 0; k0 < K / 4; ++k0) {
        int idx = k0 * 2 + half;                // float2 index: K = 4*k0 + 2*half
        v2f a = arow[idx];
        a.x *= inv;
        a.y *= inv;
        v2f b = blp[idx];
        acc = __builtin_amdgcn_wmma_f32_16x16x4_f32(
            false, a, false, b, (short)0, acc, false, false);
    }

    // ---- X @ Wr (same accumulator) ---------------------------------------
#pragma unroll
    for (int k0 = 0; k0 < K / 4; ++k0) {
        int idx = k0 * 2 + half;
        v2f a = xrow[idx];
        v2f b = brp[idx];
        acc = __builtin_amdgcn_wmma_f32_16x16x4_f32(
            false, a, false, b, (short)0, acc, false, false);
    }

    // ---- bias + activation + store ---------------------------------------
    float bv = bias[n];
    float* orow = out + (long)(tm * 16 + 8 * half) * NO + tn * 16 + row;
#pragma unroll
    for (int r = 0; r < 8; ++r) {
        float v = acc[r] + bv;
        if (RELU) v = fmaxf(v, 0.0f);
        orow[(long)r * NO] = v;
    }
}

// ---------------------------------------------------------------------------
// Launcher
// ---------------------------------------------------------------------------
extern "C" void kernel_launch(void* const* d_in, const int* in_sizes, int n_in,
                              void* d_out, int out_size, void* d_ws, size_t ws_size,
                              hipStream_t stream) {
    const float*     x   = (const float*)d_in[0];
    const long long* ei  = (const long long*)d_in[1];   // int64 edge_index [2, E]
    const float*     W1l = (const float*)d_in[2];
    const float*     W1r = (const float*)d_in[3];
    const float*     b1  = (const float*)d_in[4];
    const float*     W2l = (const float*)d_in[5];
    const float*     W2r = (const float*)d_in[6];
    const float*     b2  = (const float*)d_in[7];
    float* out = (float*)d_out;

    const long N = in_sizes[0] / 64;            // 100000, multiple of 16
    const int  E = (int)(in_sizes[1] / 2);
    const long long* src = ei;
    const long long* dst = ei + E;

    // workspace: cnt[N] | agg[N*128] (reused) | h[N*128] | 4x transposed W
    float* cnt  = (float*)d_ws;
    float* agg  = cnt + N;
    float* h    = agg + N * 128;
    float* W1lT = h + N * 128;                  // 128 x 64
    float* W1rT = W1lT + 64 * 128;              // 128 x 64
    float* W2lT = W1rT + 64 * 128;              // 64 x 128
    float* W2rT = W2lT + 128 * 64;              // 64 x 128

    const int Z = 256;

    // ---- one-time weight transposes (tiny) -------------------------------
    transpose_w<<<(64 * 128 + Z - 1) / Z, Z, 0, stream>>>(W1l, W1lT, 64, 128);
    transpose_w<<<(64 * 128 + Z - 1) / Z, Z, 0, stream>>>(W1r, W1rT, 64, 128);
    transpose_w<<<(128 * 64 + Z - 1) / Z, Z, 0, stream>>>(W2l, W2lT, 128, 64);
    transpose_w<<<(128 * 64 + Z - 1) / Z, Z, 0, stream>>>(W2r, W2rT, 128, 64);

    // ---- Layer 1 ----------------------------------------------------------
    zero_f32<<<2048, Z, 0, stream>>>(cnt, N + N * 64);          // cnt + agg(F=64)
    agg_f64<<<(E + 7) / 8, Z, 0, stream>>>(x, src, dst, agg, cnt, E);
    {
        int tiles  = (int)(N / 16) * (128 / 16);                // 50000
        int blocks = tiles / 8;                                 // exact: 6250
        sage_gemm<64, 128, true><<<blocks, Z, 0, stream>>>(
            agg, cnt, x, W1lT, W1rT, b1, h);
    }

    // ---- Layer 2 ----------------------------------------------------------
    zero_f32<<<2048, Z, 0, stream>>>(agg, N * 128);
    agg_f128<<<(E + 7) / 8, Z, 0, stream>>>(h, src, dst, agg, E);
    {
        int tiles  = (int)(N / 16) * (64 / 16);                 // 25000
        int blocks = tiles / 8;                                 // exact: 3125
        sage_gemm<128, 64, false><<<blocks, Z, 0, stream>>>(
            agg, cnt, h, W2lT, W2rT, b2, out);
    }
}